// GCN_72962904424611
// MI455X (gfx1250) — compile-verified
//
#include <hip/hip_runtime.h>
#include <hip/hip_bf16.h>
#include <math.h>

// ---------------------------------------------------------------------------
// GCN forward on MI455X (gfx1250).
// GEMMs: V_WMMA_F32_16X16X4_F32, 32x32 C macro-tile per wave (2x2 frags),
//        compile-time K/N so all loads use immediate offsets (no addr VALU).
// Edge aggregation: float4 gathers + global_atomic_add_f32 (L2-resident).
// ---------------------------------------------------------------------------

#define N_NODES 50000
#define N_EDGES 800000
#define IN_C    128
#define HID_C   128
#define EMB_C   64
#define OUT_C   16

typedef __attribute__((ext_vector_type(2))) float v2f;
typedef __attribute__((ext_vector_type(8))) float v8f;

// ---------------- degree / normalization ----------------

__global__ void gcn_deg_init(float* __restrict__ deg, int n) {
    int i = blockIdx.x * blockDim.x + threadIdx.x;
    if (i < n) deg[i] = 1.0f;   // self-loop contributes 1
}

__global__ void gcn_deg_accum(const int* __restrict__ dst, float* __restrict__ deg, int nE) {
    int i = blockIdx.x * blockDim.x + threadIdx.x;
    if (i < nE) atomicAdd(&deg[dst[i]], 1.0f);
}

__global__ void gcn_deg_inv(float* __restrict__ deg, int n) {
    int i = blockIdx.x * blockDim.x + threadIdx.x;
    if (i < n) deg[i] = rsqrtf(deg[i]);   // deg >= 1 always
}

// ---------------- WMMA GEMM: Y[M,N] = X[M,K] @ W[K,N] (+bias) ----------------
// One wave owns a (16*MR) x (16*NR) C macro-tile; K-loop unrolled in steps of 4.
// f32 16x16x4 layouts (ISA 7.12.2):
//   A 16x4 : vgpr j holds K = kBase + j + 2*half, row = lane&15
//   B 4x16 : vgpr j holds K = kBase + j + 2*half, col = lane&15
//   C/D    : vgpr q holds M = q + 8*half,        N = lane&15
template <int K, int N, int MR, int NR>
__global__ void gcn_gemm_wmma(const float* __restrict__ X, const float* __restrict__ W,
                              float* __restrict__ Y, const float* __restrict__ bias,
                              int M, int nTiles, int totalTiles) {
    int wave = blockIdx.x * (blockDim.x >> 5) + (threadIdx.x >> 5);
    if (wave >= totalTiles) return;            // uniform per-wave exit (EXEC stays full)
    int mT = wave / nTiles;
    int nT = wave - mT * nTiles;
    int m0 = mT * (16 * MR), n0 = nT * (16 * NR);
    int lane = threadIdx.x & 31;
    int lo   = lane & 15;
    int half = lane >> 4;

    // loop-invariant per-lane base pointers; all k-loop accesses are imm offsets
    const float* pA[MR];
#pragma unroll
    for (int i = 0; i < MR; ++i) {
        int r = m0 + i * 16 + lo;
        if (r >= M) r = M - 1;                 // clamp (loads only; stores predicated)
        pA[i] = X + (long long)r * K + 2 * half;
    }
    const float* pB[NR];
#pragma unroll
    for (int j = 0; j < NR; ++j)
        pB[j] = W + (long long)(2 * half) * N + (n0 + j * 16 + lo);

    v8f acc[MR][NR];
#pragma unroll
    for (int i = 0; i < MR; ++i)
#pragma unroll
        for (int j = 0; j < NR; ++j) acc[i][j] = (v8f){};

#pragma unroll
    for (int k0 = 0; k0 < K; k0 += 4) {
        v2f a[MR], b[NR];
#pragma unroll
        for (int i = 0; i < MR; ++i) {         // contiguous pair -> global_load_b64
            a[i].x = pA[i][k0];
            a[i].y = pA[i][k0 + 1];
        }
#pragma unroll
        for (int j = 0; j < NR; ++j) {         // two b32 loads, imm offsets
            b[j].x = pB[j][k0 * N];
            b[j].y = pB[j][(k0 + 1) * N];
        }
#pragma unroll
        for (int i = 0; i < MR; ++i)
#pragma unroll
            for (int j = 0; j < NR; ++j)
                acc[i][j] = __builtin_amdgcn_wmma_f32_16x16x4_f32(
                    /*neg_a=*/false, a[i], /*neg_b=*/false, b[j],
                    /*c_mod=*/(short)0, acc[i][j],
                    /*reuse_a=*/false, /*reuse_b=*/false);
    }

#pragma unroll
    for (int i = 0; i < MR; ++i) {
#pragma unroll
        for (int j = 0; j < NR; ++j) {
            int col = n0 + j * 16 + lo;
            float bv = bias ? bias[col] : 0.0f;
#pragma unroll
            for (int q = 0; q < 8; ++q) {
                int r = m0 + i * 16 + q + 8 * half;
                if (r < M) Y[(long long)r * N + col] = acc[i][j][q] + bv;
            }
        }
    }
}

// ---------------- aggregation ----------------

// agg = dinv[i]^2 * xw[i]   (self-loop term; also initializes the accumulator)
__global__ void gcn_self_init(const float* __restrict__ xw, const float* __restrict__ dinv,
                              float* __restrict__ agg, int total, int log2C) {
    int i = blockIdx.x * blockDim.x + threadIdx.x;
    if (i >= total) return;
    float di = dinv[i >> log2C];
    agg[i] = di * di * xw[i];
}

// scatter-add over edges; each thread handles one edge x 4 channels (b128 gather)
template <int C>
__global__ void gcn_edge_scatter(const int* __restrict__ src, const int* __restrict__ dst,
                                 const float* __restrict__ dinv, const float* __restrict__ xw,
                                 float* __restrict__ agg, int nEdges) {
    constexpr int GPE = C / 4;  // thread-groups per edge (power of two -> shift)
    unsigned gid = blockIdx.x * blockDim.x + threadIdx.x;
    unsigned e = gid / GPE;
    if (e >= (unsigned)nEdges) return;
    int c4 = (int)(gid - e * GPE) * 4;
    int s = src[e], d = dst[e];
    float w = dinv[s] * dinv[d];
    const float4 v = *(const float4*)(xw + (long long)s * C + c4);
    float* base = agg + (long long)d * C + c4;
    atomicAdd(base + 0, w * v.x);
    atomicAdd(base + 1, w * v.y);
    atomicAdd(base + 2, w * v.z);
    atomicAdd(base + 3, w * v.w);
}

// out = tanh(agg + bias)
__global__ void gcn_act(const float* __restrict__ agg, const float* __restrict__ bias,
                        float* __restrict__ out, int total, int cMask) {
    int i = blockIdx.x * blockDim.x + threadIdx.x;
    if (i >= total) return;
    out[i] = tanhf(agg[i] + bias[i & cMask]);
}

// ---------------------------------------------------------------------------

static inline unsigned cdiv(long long a, long long b) { return (unsigned)((a + b - 1) / b); }

extern "C" void kernel_launch(void* const* d_in, const int* in_sizes, int n_in,
                              void* d_out, int out_size, void* d_ws, size_t ws_size,
                              hipStream_t stream) {
    const float* x   = (const float*)d_in[0];
    const int*   ei  = (const int*)d_in[1];       // edge_index [2, E] row-major, int32
    const float* W1  = (const float*)d_in[2];
    const float* b1  = (const float*)d_in[3];
    const float* W2  = (const float*)d_in[4];
    const float* b2  = (const float*)d_in[5];
    const float* W3  = (const float*)d_in[6];
    const float* b3  = (const float*)d_in[7];
    const float* Wc  = (const float*)d_in[8];
    const float* bc  = (const float*)d_in[9];
    (void)n_in; (void)in_sizes; (void)ws_size; (void)out_size;

    const int* src = ei;
    const int* dst = ei + N_EDGES;

    float* out  = (float*)d_out;                       // [N, 16]
    float* hout = out + (long long)N_NODES * OUT_C;    // [N, 64]  (second output)

    // workspace layout
    float* ws   = (float*)d_ws;
    float* dinv = ws;                                  // 50000   (deg -> dinv in place)
    float* xw   = ws + 50048;                          // 50000*128
    float* agg  = xw + (long long)N_NODES * HID_C;     // 50000*128
    float* hbuf = agg + (long long)N_NODES * HID_C;    // 50000*128

    const int TPB = 256;                               // 8 waves per block
    const int mTiles32 = (N_NODES + 31) / 32;          // 1563 (32-row macro-tiles)

    // ---- degrees / symmetric normalization ----
    gcn_deg_init <<<cdiv(N_NODES, TPB), TPB, 0, stream>>>(dinv, N_NODES);
    gcn_deg_accum<<<cdiv(N_EDGES, TPB), TPB, 0, stream>>>(dst, dinv, N_EDGES);
    gcn_deg_inv  <<<cdiv(N_NODES, TPB), TPB, 0, stream>>>(dinv, N_NODES);

    // ---- layer 1: 128 -> 128 ----
    {
        int nT = HID_C / 32, tot = mTiles32 * nT;      // 4 * 1563
        gcn_gemm_wmma<IN_C, HID_C, 2, 2><<<cdiv(tot, 8), TPB, 0, stream>>>(
            x, W1, xw, nullptr, N_NODES, nT, tot);
        int total = N_NODES * HID_C;
        gcn_self_init<<<cdiv(total, TPB), TPB, 0, stream>>>(xw, dinv, agg, total, 7);
        gcn_edge_scatter<HID_C><<<cdiv((long long)N_EDGES * (HID_C / 4), TPB), TPB, 0, stream>>>(
            src, dst, dinv, xw, agg, N_EDGES);
        gcn_act<<<cdiv(total, TPB), TPB, 0, stream>>>(agg, b1, hbuf, total, HID_C - 1);
    }

    // ---- layer 2: 128 -> 128 ----
    {
        int nT = HID_C / 32, tot = mTiles32 * nT;
        gcn_gemm_wmma<HID_C, HID_C, 2, 2><<<cdiv(tot, 8), TPB, 0, stream>>>(
            hbuf, W2, xw, nullptr, N_NODES, nT, tot);
        int total = N_NODES * HID_C;
        gcn_self_init<<<cdiv(total, TPB), TPB, 0, stream>>>(xw, dinv, agg, total, 7);
        gcn_edge_scatter<HID_C><<<cdiv((long long)N_EDGES * (HID_C / 4), TPB), TPB, 0, stream>>>(
            src, dst, dinv, xw, agg, N_EDGES);
        gcn_act<<<cdiv(total, TPB), TPB, 0, stream>>>(agg, b2, hbuf, total, HID_C - 1);
    }

    // ---- layer 3: 128 -> 64, activations written straight into d_out's h region ----
    {
        int nT = EMB_C / 32, tot = mTiles32 * nT;      // 2 * 1563
        gcn_gemm_wmma<HID_C, EMB_C, 2, 2><<<cdiv(tot, 8), TPB, 0, stream>>>(
            hbuf, W3, xw, nullptr, N_NODES, nT, tot);
        int total = N_NODES * EMB_C;
        gcn_self_init<<<cdiv(total, TPB), TPB, 0, stream>>>(xw, dinv, agg, total, 6);
        gcn_edge_scatter<EMB_C><<<cdiv((long long)N_EDGES * (EMB_C / 4), TPB), TPB, 0, stream>>>(
            src, dst, dinv, xw, agg, N_EDGES);
        gcn_act<<<cdiv(total, TPB), TPB, 0, stream>>>(agg, b3, hout, total, EMB_C - 1);
    }

    // ---- classifier: out = h @ Wc + bc  (64 -> 16), 32x16 macro-tile ----
    {
        int nT = 1, tot = mTiles32;                    // single 16-col tile
        gcn_gemm_wmma<EMB_C, OUT_C, 2, 1><<<cdiv(tot, 8), TPB, 0, stream>>>(
            hout, Wc, out, bc, N_NODES, nT, tot);
    }
}